// CrowdCountingLoss_74363063763130
// MI455X (gfx1250) — compile-verified
//
#include <hip/hip_runtime.h>
#include <hip/hip_bf16.h>
#include <math.h>

// ---------------------------------------------------------------------------
// CrowdCountingLoss for MI455X (gfx1250, wave32)
//   density = mean((pred-gtb)^2); count = |sum(pred)-sum(gt)|
//   spatial = debiased unbalanced Sinkhorn divergence (p=2, blur=.05, reach=.5)
// Cost matrices built with V_WMMA_F32_16X16X4_F32 (full f32 precision, needed
// because values are exponentiated at 1/eps = 400x). All four matrices
// (Cxy, Cxx, Cyy, Cyx) are stored row-major so every Sinkhorn sweep is a
// coalesced row read of L2-resident data (192MB L2 >> 9.5MB working set).
// ---------------------------------------------------------------------------

typedef float v2f __attribute__((ext_vector_type(2)));
typedef float v8f __attribute__((ext_vector_type(8)));

#define N_PTS   768
#define N2      (768 * 768)      // 589824
#define EPS     0.0025f          // blur^2
#define INV_EPS 400.0f
#define RHO     0.25f            // reach^2
#define INV_RHO 4.0f
#define LAM     (0.25f / 0.2525f)          // rho / (rho + eps)
#define LOGW    (-6.643789733147672f)      // -log(768)
#define SCALE_  0.25125f                   // rho + eps/2
#define NBLK_R  576                        // reduction blocks (576*1024 = N2)

// ---------------- wave/block reductions (wave32) ---------------------------
__device__ __forceinline__ float waveSum(float v) {
    #pragma unroll
    for (int o = 16; o > 0; o >>= 1) v += __shfl_xor(v, o, 32);
    return v;
}
__device__ __forceinline__ float waveMax(float v) {
    #pragma unroll
    for (int o = 16; o > 0; o >>= 1) v = fmaxf(v, __shfl_xor(v, o, 32));
    return v;
}
__device__ __forceinline__ float blockSum(float v, float* sm) {
    int wid = threadIdx.x >> 5, lid = threadIdx.x & 31;
    v = waveSum(v);
    __syncthreads();
    if (lid == 0) sm[wid] = v;
    __syncthreads();
    if (wid == 0) {
        float t = (lid < (int)(blockDim.x >> 5)) ? sm[lid] : 0.0f;
        t = waveSum(t);
        if (lid == 0) sm[0] = t;
    }
    __syncthreads();
    return sm[0];
}
__device__ __forceinline__ float blockMax(float v, float* sm) {
    int wid = threadIdx.x >> 5, lid = threadIdx.x & 31;
    v = waveMax(v);
    __syncthreads();
    if (lid == 0) sm[wid] = v;
    __syncthreads();
    if (wid == 0) {
        float t = (lid < (int)(blockDim.x >> 5)) ? sm[lid] : -3.0e38f;
        t = waveMax(t);
        if (lid == 0) sm[0] = t;
    }
    __syncthreads();
    return sm[0];
}

// ---------------- init: zero sinkhorn potentials ---------------------------
__global__ void k_init_pot(float* pot0) {
    int i = blockIdx.x * blockDim.x + threadIdx.x;
    if (i < 4 * N_PTS) pot0[i] = 0.0f;
}

// ---------------- per-row squared norms ------------------------------------
__global__ void k_row_norms(const float* __restrict__ x, const float* __restrict__ y,
                            float* __restrict__ x2, float* __restrict__ y2) {
    __shared__ float sm[8];
    const float* src = (blockIdx.y == 0) ? x : y;
    float*       dst = (blockIdx.y == 0) ? x2 : y2;
    int row = blockIdx.x;
    float s = 0.0f;
    for (int j = threadIdx.x; j < N_PTS; j += 256) {
        float v = src[row * N_PTS + j];
        s += v * v;
    }
    s = blockSum(s, sm);
    if (threadIdx.x == 0) dst[row] = s;
}

// ---------------- density / count partial reductions -----------------------
__global__ void k_reduce_inputs(const float* __restrict__ pred,
                                const float* __restrict__ gt,
                                const float* __restrict__ gtb,
                                float* __restrict__ part) {
    __shared__ float sm[8];
    int base = blockIdx.x * 1024;
    float d = 0.0f, sp = 0.0f, sg = 0.0f;
    for (int t = threadIdx.x; t < 1024; t += 256) {
        int i = base + t;
        float pv = pred[i];
        float df = pv - gtb[i];
        d  += df * df;
        sp += pv;
        sg += gt[i];
    }
    d  = blockSum(d, sm);
    sp = blockSum(sp, sm);
    sg = blockSum(sg, sm);
    if (threadIdx.x == 0) {
        part[blockIdx.x]              = d;
        part[NBLK_R + blockIdx.x]     = sp;
        part[2 * NBLK_R + blockIdx.x] = sg;
    }
}

// ---------------- cost-matrix GEMM via V_WMMA_F32_16X16X4_F32 --------------
// Stores Cs = C/eps = 200 * max(|a|^2 + |b|^2 - 2 a.b, 0)
// blockIdx.z: 0 -> Cxy (x,y)  1 -> Cxx (x,x)  2 -> Cyy (y,y)  3 -> Cyx (y,x)
__global__ void k_gemm_cost(const float* __restrict__ x, const float* __restrict__ y,
                            const float* __restrict__ x2, const float* __restrict__ y2,
                            float* __restrict__ Cs) {
    const int mat = blockIdx.z;
    const float* A;  const float* B;  const float* r2A; const float* r2B; float* out;
    if (mat == 0)      { A = x; B = y; r2A = x2; r2B = y2; out = Cs; }
    else if (mat == 1) { A = x; B = x; r2A = x2; r2B = x2; out = Cs + N2; }
    else if (mat == 2) { A = y; B = y; r2A = y2; r2B = y2; out = Cs + 2 * N2; }
    else               { A = y; B = x; r2A = y2; r2B = x2; out = Cs + 3 * N2; }

    const int lane = threadIdx.x & 31;
    const int mrow = lane & 15;            // M (for A) / N (for B) within tile
    const int koff = (lane >> 4) * 2;      // lanes 16-31 hold K=2,3 of each step

    const float* aptr = A + (blockIdx.x * 16 + mrow) * N_PTS + koff;
    const float* bptr = B + (blockIdx.y * 16 + mrow) * N_PTS + koff;

    v8f c = {};
    #pragma unroll 8
    for (int k = 0; k < N_PTS; k += 4) {
        v2f a = *(const v2f*)(aptr + k);
        v2f b = *(const v2f*)(bptr + k);
        // D = A(16x4,f32) x B(4x16,f32) + C ; emits v_wmma_f32_16x16x4_f32
        c = __builtin_amdgcn_wmma_f32_16x16x4_f32(
                /*neg_a=*/false, a, /*neg_b=*/false, b,
                /*c_mod=*/(short)0, c, /*reuse_a=*/false, /*reuse_b=*/false);
    }

    // C/D layout: lanes 0-15 -> M = r, lanes 16-31 -> M = 8 + r ; N = lane&15
    const int rowBase = blockIdx.x * 16 + ((lane >> 4) << 3);
    const int col     = blockIdx.y * 16 + (lane & 15);
    const float yc = r2B[col];
    #pragma unroll
    for (int r = 0; r < 8; ++r) {
        int row  = rowBase + r;
        float d2 = r2A[row] + yc - 2.0f * c[r];
        float v  = d2 > 0.0f ? d2 * 200.0f : 0.0f;   // (0.5*d2) * (1/eps)
        out[row * N_PTS + col] = v;
    }
}

// ---------------- one damped symmetric Sinkhorn sweep ----------------------
// blockIdx.y = problem: 0:f<-g (Cxy) 1:g<-f (Cyx) 2:p (Cxx) 3:q (Cyy)
// All reads are coalesced row reads (Cyx stored explicitly).
__global__ void k_sinkhorn_iter(const float* __restrict__ Cs,
                                const float* __restrict__ potOld,
                                float* __restrict__ dst, int isFinal) {
    __shared__ float sm[8];
    const int row  = blockIdx.x;
    const int prob = blockIdx.y;

    const float* srcPot;
    const float* Crow;
    if (prob == 0)      { srcPot = potOld + N_PTS;     Crow = Cs + row * N_PTS; }          // f <- g
    else if (prob == 1) { srcPot = potOld;             Crow = Cs + 3 * N2 + row * N_PTS; } // g <- f
    else if (prob == 2) { srcPot = potOld + 2 * N_PTS; Crow = Cs + N2 + row * N_PTS; }    // p
    else                { srcPot = potOld + 3 * N_PTS; Crow = Cs + 2 * N2 + row * N_PTS; }// q

    float vals[3];
    float m = -3.0e38f;
    #pragma unroll
    for (int t = 0; t < 3; ++t) {
        int j = threadIdx.x + t * 256;
        float v = LOGW + srcPot[j] * INV_EPS - Crow[j];   // h_j - C_ij/eps
        vals[t] = v;
        m = fmaxf(m, v);
    }
    m = blockMax(m, sm);
    float s = 0.0f;
    #pragma unroll
    for (int t = 0; t < 3; ++t) s += __expf(vals[t] - m);
    s = blockSum(s, sm);

    if (threadIdx.x == 0) {
        float softmin = -EPS * (m + __logf(s));
        float r = LAM * softmin;
        float outv = isFinal ? r
                             : 0.5f * (potOld[prob * N_PTS + row] + r);
        dst[prob * N_PTS + row] = outv;
    }
}

// ---------------- final combine --------------------------------------------
__global__ void k_finalize(const float* __restrict__ part,
                           const float* __restrict__ pot,       // buf after 30 iters
                           const float* __restrict__ fgFinal,   // extrapolated f,g
                           float* __restrict__ out) {
    __shared__ float sm[8];
    float d = 0.0f, sp = 0.0f, sg = 0.0f;
    for (int i = threadIdx.x; i < NBLK_R; i += 256) {
        d  += part[i];
        sp += part[NBLK_R + i];
        sg += part[2 * NBLK_R + i];
    }
    d  = blockSum(d, sm);
    sp = blockSum(sp, sm);
    sg = blockSum(sg, sm);

    float ta = 0.0f, tb = 0.0f;
    for (int i = threadIdx.x; i < N_PTS; i += 256) {
        float f = fgFinal[i];
        float g = fgFinal[N_PTS + i];
        float p = pot[2 * N_PTS + i];
        float q = pot[3 * N_PTS + i];
        ta += expf(-p * INV_RHO) - expf(-f * INV_RHO);
        tb += expf(-q * INV_RHO) - expf(-g * INV_RHO);
    }
    ta = blockSum(ta, sm);
    tb = blockSum(tb, sm);

    if (threadIdx.x == 0) {
        const float invN = 1.0f / (float)N_PTS;
        float spatial = SCALE_ * (invN * ta + invN * tb);
        float density = d * (1.0f / (float)N2);
        float count   = fabsf(sp - sg);
        out[0] = density + count + spatial;   // ALPHA = 1
    }
}

// ---------------------------------------------------------------------------
extern "C" void kernel_launch(void* const* d_in, const int* in_sizes, int n_in,
                              void* d_out, int out_size, void* d_ws, size_t ws_size,
                              hipStream_t stream) {
    const float* pred = (const float*)d_in[0];   // (768,768)
    const float* gt   = (const float*)d_in[1];   // (1,1,768,768)
    const float* gtb  = (const float*)d_in[2];   // (1,1,768,768)
    float* out = (float*)d_out;

    float* ws   = (float*)d_ws;
    float* Cs   = ws;                       // 4 * N2 floats (Cxy, Cxx, Cyy, Cyx) as C/eps
    float* x2   = Cs + 4 * N2;              // 768
    float* y2   = x2 + N_PTS;               // 768
    float* pot0 = y2 + N_PTS;               // 4*768
    float* pot1 = pot0 + 4 * N_PTS;         // 4*768
    float* fgF  = pot1 + 4 * N_PTS;         // 2*768
    float* part = fgF + 2 * N_PTS;          // 3*NBLK_R

    // 1) zero potentials (f,g,p,q)
    k_init_pot<<<dim3((4 * N_PTS + 255) / 256), dim3(256), 0, stream>>>(pot0);

    // 2) row squared norms for x=pred, y=gt
    k_row_norms<<<dim3(N_PTS, 2), dim3(256), 0, stream>>>(pred, gt, x2, y2);

    // 3) density / count partials
    k_reduce_inputs<<<dim3(NBLK_R), dim3(256), 0, stream>>>(pred, gt, gtb, part);

    // 4) cost matrices via f32 WMMA (48x48 tiles of 16x16, 4 matrices)
    k_gemm_cost<<<dim3(48, 48, 4), dim3(32), 0, stream>>>(pred, gt, x2, y2, Cs);

    // 5) 30 damped Jacobi Sinkhorn sweeps (ping-pong)
    float* cur = pot0;
    float* nxt = pot1;
    for (int it = 0; it < 30; ++it) {
        k_sinkhorn_iter<<<dim3(N_PTS, 4), dim3(256), 0, stream>>>(Cs, cur, nxt, 0);
        float* t = cur; cur = nxt; nxt = t;
    }
    // after 30 iters result is back in pot0 (cur == pot0)

    // 6) final non-averaged extrapolation for f,g only
    k_sinkhorn_iter<<<dim3(N_PTS, 2), dim3(256), 0, stream>>>(Cs, cur, fgF, 1);

    // 7) combine everything into the scalar loss
    k_finalize<<<dim3(1), dim3(256), 0, stream>>>(part, cur, fgF, out);
}